// RandomElasticDeformation3D_54949811585243
// MI455X (gfx1250) — compile-verified
//
#include <hip/hip_runtime.h>
#include <math.h>

// ---------------------------------------------------------------------------
// RandomElasticDeformation3D for MI455X (gfx1250, wave32)
//
// flow[b,d,h,w,c] = ALPHA * sum_i A[d,i] * Q[hw][(b*4+i)*3+c]
//   Q[hw][n]      = sum_jk (A[h,j]*A[w,k]) * coarseB[jk][n]       <-- WMMA GEMM
//   A (128x4)     = Gauss16 conv (pad 7,8) of bicubic resize matrix M (128x4)
//
// ws layout (floats): [0,512)  A ; [512,1024) coarseB 16x32 (cols>=24 zero)
//                     [1024, 1024+16384*24) Q
// ---------------------------------------------------------------------------

typedef __attribute__((ext_vector_type(2))) float v2f;
typedef __attribute__((ext_vector_type(8))) float v8f;

#define NVOX (2 * 128 * 128 * 128)

__device__ __forceinline__ int clampi(int v, int lo, int hi) {
    return v < lo ? lo : (v > hi ? hi : v);
}

// ---- Kernel 1: build A (=Gauss*resize) and rearranged coarseB -------------
__global__ __launch_bounds__(128) void elastic_prep_kernel(
    const float* __restrict__ coarse, float* __restrict__ ws) {
    __shared__ float Mlds[128][4];
    const int t = threadIdx.x;  // 0..127

    // bicubic resize row for output position t (n_in=4, n_out=128)
    const float a = -0.75f;
    float src = (t + 0.5f) * (4.0f / 128.0f) - 0.5f;
    float f = floorf(src);
    float tt = src - f;
    float row[4] = {0.f, 0.f, 0.f, 0.f};
#pragma unroll
    for (int o = 0; o < 4; ++o) {
        float x = tt - (float)(o - 1);
        float ax = fabsf(x);
        float wv;
        if (ax <= 1.0f)
            wv = ((a + 2.0f) * ax - (a + 3.0f)) * ax * ax + 1.0f;
        else if (ax < 2.0f)
            wv = (((ax - 5.0f) * ax + 8.0f) * ax - 4.0f) * a;
        else
            wv = 0.0f;
        int idx = clampi((int)f + (o - 1), 0, 3);
        row[idx] += wv;  // matches .at[].add with clipped indices
    }
#pragma unroll
    for (int i = 0; i < 4; ++i) Mlds[t][i] = row[i];
    __syncthreads();

    // 16-tap gaussian: taps at offsets -7..8, sigma=2.5
    float g[16];
    float gs = 0.f;
#pragma unroll
    for (int m = 0; m < 16; ++m) {
        float axm = (float)(m - 7);
        g[m] = expf(-(axm * axm) / 12.5f);
        gs += g[m];
    }
    float inv = 1.0f / gs;

    // A[t][i] = sum_m g[m] * M[t + m - 7][i], zero padded
    float acc[4] = {0.f, 0.f, 0.f, 0.f};
#pragma unroll
    for (int m = 0; m < 16; ++m) {
        int s = t + m - 7;
        if (s >= 0 && s < 128) {
            float gm = g[m] * inv;
#pragma unroll
            for (int i = 0; i < 4; ++i) acc[i] += gm * Mlds[s][i];
        }
    }
#pragma unroll
    for (int i = 0; i < 4; ++i) ws[t * 4 + i] = acc[i];

    // coarseB[jk][n] (16x32, n=(b*4+i)*3+c valid for n<24, else 0)
    for (int s = t; s < 512; s += 128) {
        int jk = s >> 5;
        int n = s & 31;
        float v = 0.f;
        if (n < 24) {
            int b = n / 12;
            int rem = n % 12;
            int i = rem / 3;
            int c = rem % 3;
            int j = jk >> 2;
            int k = jk & 3;
            v = coarse[((((b * 4 + i) * 4 + j) * 4 + k) * 3) + c];
        }
        ws[512 + s] = v;
    }
}

// ---- Kernel 2: Q = P x coarseB via V_WMMA_F32_16X16X4_F32 -----------------
// one wave -> one 16x16 output tile; 1024 M-tiles x 2 N-tiles = 2048 waves
__global__ __launch_bounds__(256) void elastic_flowgemm_kernel(
    float* __restrict__ ws) {
    const float* A = ws;           // 128x4
    const float* CB = ws + 512;    // 16x32
    float* Q = ws + 1024;          // 16384x24

    int wid = (int)((blockIdx.x * blockDim.x + threadIdx.x) >> 5);
    int lane = threadIdx.x & 31;
    int gsel = lane >> 4;   // lane group: 0 -> K{0,1}, 1 -> K{2,3}
    int col = lane & 15;

    int mt = wid >> 1;      // hw tile (0..1023)
    int nt = wid & 1;       // n tile  (0..1)

    // A-fragment row (same rows in both lane halves)
    int hwRow = mt * 16 + col;
    int h = hwRow >> 7;
    int w = hwRow & 127;

    v8f c = {};
#pragma unroll
    for (int kk = 0; kk < 4; ++kk) {
        v2f av, bv;
#pragma unroll
        for (int v = 0; v < 2; ++v) {
            int jk = kk * 4 + 2 * gsel + v;   // global K index
            int j = jk >> 2;
            int k = jk & 3;
            av[v] = A[h * 4 + j] * A[w * 4 + k];       // P[hw][jk] on the fly
            bv[v] = CB[jk * 32 + nt * 16 + col];       // B[k][n]
        }
        c = __builtin_amdgcn_wmma_f32_16x16x4_f32(
            /*neg_a=*/false, av, /*neg_b=*/false, bv,
            /*c_mod=*/(short)0, c, /*reuse_a=*/false, /*reuse_b=*/false);
    }

    int n = nt * 16 + col;
    if (n < 24) {
#pragma unroll
        for (int r = 0; r < 8; ++r) {
            int rowOut = mt * 16 + r + gsel * 8;  // C: VGPR r -> rows r / r+8
            Q[rowOut * 24 + n] = c[r];
        }
    }
}

// ---- Kernel 3: finish d-contraction, warp, trilinear + nearest sample -----
__global__ __launch_bounds__(256) void elastic_warp_kernel(
    const float* __restrict__ image, const float* __restrict__ label,
    const float* __restrict__ ws, float* __restrict__ outImg,
    float* __restrict__ outLab) {
    int idx = (int)(blockIdx.x * blockDim.x + threadIdx.x);
    if (idx >= NVOX) return;

    int w = idx & 127;
    int h = (idx >> 7) & 127;
    int d = (idx >> 14) & 127;
    int b = idx >> 21;
    int hw = h * 128 + w;

    const float* A = ws;
    const float* q = ws + 1024 + hw * 24;

    float ad0 = A[d * 4 + 0], ad1 = A[d * 4 + 1];
    float ad2 = A[d * 4 + 2], ad3 = A[d * 4 + 3];

    float fl[3];
#pragma unroll
    for (int cc = 0; cc < 3; ++cc) {
        float s = ad0 * q[0 * 3 + cc] + ad1 * q[3 + cc] +
                  ad2 * q[6 + cc] + ad3 * q[9 + cc];
        // q rows are (b*4+i); offset by batch
        if (b) {
            s = ad0 * q[12 + cc] + ad1 * q[15 + cc] +
                ad2 * q[18 + cc] + ad3 * q[21 + cc];
        }
        fl[cc] = 35.0f * s;  // ALPHA
    }

    float wd = (float)d + fl[0];
    float wh = (float)h + fl[1];
    float ww = (float)w + fl[2];

    float fd = floorf(wd), fh = floorf(wh), fw = floorf(ww);
    float td = wd - fd, th = wh - fh, tw = ww - fw;

    int d0 = clampi((int)fd, 0, 127), d1 = clampi((int)fd + 1, 0, 127);
    int h0 = clampi((int)fh, 0, 127), h1 = clampi((int)fh + 1, 0, 127);
    int w0 = clampi((int)fw, 0, 127), w1 = clampi((int)fw + 1, 0, 127);

    const float2* img2 = (const float2*)image + (size_t)b * (128 * 128 * 128);
#define G2(di, hi, wi) img2[(((di) * 128 + (hi)) * 128 + (wi))]
    float2 v000 = G2(d0, h0, w0), v001 = G2(d0, h0, w1);
    float2 v010 = G2(d0, h1, w0), v011 = G2(d0, h1, w1);
    float2 v100 = G2(d1, h0, w0), v101 = G2(d1, h0, w1);
    float2 v110 = G2(d1, h1, w0), v111 = G2(d1, h1, w1);
#undef G2

    float otw = 1.0f - tw, oth = 1.0f - th, otd = 1.0f - td;
#pragma unroll
    for (int c = 0; c < 2; ++c) {
        float a000 = c ? v000.y : v000.x, a001 = c ? v001.y : v001.x;
        float a010 = c ? v010.y : v010.x, a011 = c ? v011.y : v011.x;
        float a100 = c ? v100.y : v100.x, a101 = c ? v101.y : v101.x;
        float a110 = c ? v110.y : v110.x, a111 = c ? v111.y : v111.x;
        float c00 = a000 * otw + a001 * tw;
        float c01 = a010 * otw + a011 * tw;
        float c10 = a100 * otw + a101 * tw;
        float c11 = a110 * otw + a111 * tw;
        float c0 = c00 * oth + c01 * th;
        float c1 = c10 * oth + c11 * th;
        outImg[(size_t)idx * 2 + c] = c0 * otd + c1 * td;
    }

    // nearest (round half to even, like jnp.round) for label
    int nd = clampi((int)rintf(wd), 0, 127);
    int nh = clampi((int)rintf(wh), 0, 127);
    int nw = clampi((int)rintf(ww), 0, 127);
    const float* lab = label + (size_t)b * (128 * 128 * 128);
    outLab[idx] = lab[((nd * 128 + nh) * 128 + nw)];
}

extern "C" void kernel_launch(void* const* d_in, const int* in_sizes, int n_in,
                              void* d_out, int out_size, void* d_ws,
                              size_t ws_size, hipStream_t stream) {
    const float* image = (const float*)d_in[0];        // (2,128,128,128,2)
    const float* label = (const float*)d_in[1];        // (2,128,128,128,1)
    const float* coarse = (const float*)d_in[2];       // (2,4,4,4,3)
    float* ws = (float*)d_ws;
    float* outImg = (float*)d_out;                     // 8388608 floats
    float* outLab = (float*)d_out + (size_t)NVOX * 2;  // 4194304 floats

    elastic_prep_kernel<<<1, 128, 0, stream>>>(coarse, ws);
    // 2048 waves * 32 lanes / 256 threads = 256 blocks
    elastic_flowgemm_kernel<<<256, 256, 0, stream>>>(ws);
    elastic_warp_kernel<<<(NVOX + 255) / 256, 256, 0, stream>>>(
        image, label, ws, outImg, outLab);
}